// PhobiaLoss_87634512707975
// MI455X (gfx1250) — compile-verified
//
#include <hip/hip_runtime.h>
#include <math.h>

// Problem constants from the reference: N=8192, S=13, B=2, C=3
#define SS    169     // S*S
#define CH_P  16      // B*(5+C) prediction channels  (64 bytes/cell, 16B aligned)
#define CH_T  14      // 4 + 1 + 5 + C + 1 target channels (56 bytes/cell, 8B aligned)
#define EPSV  1e-6f

#define L_COORD 5.0f
#define L_OBJ   1.0f
#define L_NOOBJ 0.5f
#define L_CLASS 1.0f

typedef float v2f __attribute__((ext_vector_type(2)));
typedef float v8f __attribute__((ext_vector_type(8)));

// Full 32-lane (wave32) sum using the matrix pipe.
// A (16x4 f32): a0 = v  -> A[m,0]=v_m (lanes 0..15), A[m,2]=v_{m+16} (lanes 16..31)
//               a1 = 0  -> A[m,1]=A[m,3]=0
// B (4x16 f32) = ones   -> D[m,n] = v_m + v_{m+16}   (all columns identical)
// D VGPR j holds s_j in lanes 0..15 and s_{j+8} in lanes 16..31, so
// sum(d0..d7) gives half-sums per 16-lane group; one xor-16 shuffle finishes.
// Requires EXEC all-1s: only call from convergent code.
__device__ __forceinline__ float wave_sum32(float v) {
    v2f a; a[0] = v;    a[1] = 0.0f;
    v2f b; b[0] = 1.0f; b[1] = 1.0f;
    v8f c = {};
    c = __builtin_amdgcn_wmma_f32_16x16x4_f32(
            /*neg_a=*/false, a, /*neg_b=*/false, b,
            /*c_mod=*/(short)0, c, /*reuse_a=*/false, /*reuse_b=*/false);
    float t = ((c[0] + c[1]) + (c[2] + c[3])) + ((c[4] + c[5]) + (c[6] + c[7]));
    t += __shfl_xor(t, 16, 32);
    return t;
}

// log(sigmoid(z)), numerically stable (matches jax.nn.log_sigmoid)
__device__ __forceinline__ float logsig(float z) {
    return fminf(z, 0.0f) - log1pf(expf(-fabsf(z)));
}

__global__ void yolo_loss_partial(const float* __restrict__ pred,
                                  const float* __restrict__ targ,
                                  float* __restrict__ part,
                                  int total_cells) {
    const int tid    = blockIdx.x * blockDim.x + threadIdx.x;
    const int stride = gridDim.x * blockDim.x;

    float acc_coord = 0.f, acc_obj = 0.f, acc_noobj = 0.f, acc_cls = 0.f, acc_n = 0.f;

    for (int i = tid; i < total_cells; i += stride) {
        // predictions: 16 floats, 64B aligned -> 4x global_load_b128
        const float4* pb = (const float4*)(pred + (size_t)i * CH_P);
        float4 p0 = pb[0];   // ch 0..3   (x,y,w,h of box 0)
        float4 p1 = pb[1];   // ch 4..7   (.x = obj logit box 0)
        float4 p2 = pb[2];   // ch 8..11  (.y = obj logit box 1, .z/.w = logits 0,1)
        float4 p3 = pb[3];   // ch 12..15 (.x = logit 2)

        // targets: 14 floats, base 8B aligned -> b64/b32 loads
        const float* tb = targ + (size_t)i * CH_T;
        float2 t01 = *(const float2*)(tb + 0);    // tx, ty
        float2 t23 = *(const float2*)(tb + 2);    // tw, th
        float  tcf = tb[4];                       // conf (0 or 1)
        float2 tab = *(const float2*)(tb + 10);   // onehot[0], onehot[1]
        float  tcc = tb[12];                      // onehot[2]

        float m = (tcf > 0.0f) ? 1.0f : 0.0f;

        // ---- coord ----
        float px = 1.0f / (1.0f + expf(-p0.x));
        float py = 1.0f / (1.0f + expf(-p0.y));
        float dx = px - t01.x, dy = py - t01.y;
        float sw = sqrtf(fabsf(p0.z) + EPSV) - sqrtf(fabsf(t23.x) + EPSV);
        float sh = sqrtf(fabsf(p0.w) + EPSV) - sqrtf(fabsf(t23.y) + EPSV);
        acc_coord += m * (dx * dx + dy * dy + sw * sw + sh * sh);

        // ---- objectness BCE (channel 4) ----
        float z   = p1.x;
        float bce = -(tcf * logsig(z) + (1.0f - tcf) * logsig(-z));
        acc_obj  += m * bce;

        // ---- no-object: softplus of channels 4 and 9 ----
        float sp4 = -logsig(-z);
        float sp9 = -logsig(-p2.y);
        acc_noobj += (1.0f - m) * (sp4 + sp9);

        // ---- class cross-entropy (logits ch 10..12) ----
        float l0 = p2.z, l1 = p2.w, l2 = p3.x;
        float mx  = fmaxf(l0, fmaxf(l1, l2));
        float lse = mx + logf(expf(l0 - mx) + expf(l1 - mx) + expf(l2 - mx));
        // first-max argmax over the one-hot row (matches jnp.argmax tie rules)
        float a = tab.x, b = tab.y, cc = tcc;
        float tl = (a >= b) ? ((a >= cc) ? l0 : l2) : ((b >= cc) ? l1 : l2);
        acc_cls += m * (lse - tl);

        acc_n += m;
    }

    // convergent here: WMMA wave reduction (EXEC all-1s)
    float wcoord = wave_sum32(acc_coord);
    float wobj   = wave_sum32(acc_obj);
    float wnoobj = wave_sum32(acc_noobj);
    float wcls   = wave_sum32(acc_cls);
    float wn     = wave_sum32(acc_n);

    __shared__ float s[8][5];
    const int lane = threadIdx.x & 31;
    const int wid  = threadIdx.x >> 5;
    if (lane == 0) {
        s[wid][0] = wcoord; s[wid][1] = wobj; s[wid][2] = wnoobj;
        s[wid][3] = wcls;   s[wid][4] = wn;
    }
    __syncthreads();
    if (threadIdx.x == 0) {
        float r0 = 0.f, r1 = 0.f, r2 = 0.f, r3 = 0.f, r4 = 0.f;
        for (int w = 0; w < 8; ++w) {
            r0 += s[w][0]; r1 += s[w][1]; r2 += s[w][2]; r3 += s[w][3]; r4 += s[w][4];
        }
        float* o = part + (size_t)blockIdx.x * 5;
        o[0] = r0; o[1] = r1; o[2] = r2; o[3] = r3; o[4] = r4;
    }
}

__global__ void yolo_loss_final(const float* __restrict__ part, int nblocks,
                                float* __restrict__ out,
                                float inv_batch, float total_cells) {
    float r0 = 0.f, r1 = 0.f, r2 = 0.f, r3 = 0.f, r4 = 0.f;
    for (int i = threadIdx.x; i < nblocks; i += blockDim.x) {
        const float* p = part + (size_t)i * 5;
        r0 += p[0]; r1 += p[1]; r2 += p[2]; r3 += p[3]; r4 += p[4];
    }
    r0 = wave_sum32(r0);
    r1 = wave_sum32(r1);
    r2 = wave_sum32(r2);
    r3 = wave_sum32(r3);
    r4 = wave_sum32(r4);

    __shared__ float s[8][5];
    const int lane = threadIdx.x & 31;
    const int wid  = threadIdx.x >> 5;
    if (lane == 0) {
        s[wid][0] = r0; s[wid][1] = r1; s[wid][2] = r2; s[wid][3] = r3; s[wid][4] = r4;
    }
    __syncthreads();
    if (threadIdx.x == 0) {
        float coord = 0.f, obj = 0.f, noobj = 0.f, cls = 0.f, nobj = 0.f;
        for (int w = 0; w < 8; ++w) {
            coord += s[w][0]; obj += s[w][1]; noobj += s[w][2];
            cls   += s[w][3]; nobj += s[w][4];
        }
        float div_obj   = (nobj > 0.0f) ? inv_batch : 1.0f;
        float n_noobj   = total_cells - nobj;
        float div_noobj = (n_noobj > 0.0f) ? inv_batch : 1.0f;
        out[0] = L_COORD * coord * div_obj
               + L_OBJ   * obj   * div_obj
               + L_NOOBJ * noobj * div_noobj
               + L_CLASS * cls   * div_obj;
    }
}

extern "C" void kernel_launch(void* const* d_in, const int* in_sizes, int n_in,
                              void* d_out, int out_size, void* d_ws, size_t ws_size,
                              hipStream_t stream) {
    const float* pred = (const float*)d_in[0];
    const float* targ = (const float*)d_in[1];

    const int total_cells = in_sizes[0] / CH_P;   // N * S * S
    const int batch       = total_cells / SS;     // N

    float* part = (float*)d_ws;                   // nblocks * 5 floats (20 KB)
    const int nblocks = 1024;
    const int nthreads = 256;                     // 8 wave32s per block

    yolo_loss_partial<<<nblocks, nthreads, 0, stream>>>(pred, targ, part, total_cells);
    yolo_loss_final<<<1, nthreads, 0, stream>>>(part, nblocks, (float*)d_out,
                                                1.0f / (float)batch, (float)total_cells);
}